// BraggNN_10677288698158
// MI455X (gfx1250) — compile-verified
//
#include <hip/hip_runtime.h>

// ---------------------------------------------------------------------------
// CDNA5 (gfx1250) fp32 WMMA implementation of the BraggNN forward pass.
// All GEMM-shaped ops use V_WMMA_F32_16X16X4_F32 (wave32, 16x16 tiles).
//
// Weights are pre-transposed once per launch into zero-padded [K][N] layout
// so that every B-fragment load is unconditional and lane-coalesced, and no
// predicated (branchy) loads appear inside WMMA loops.
//
// INVARIANT: the launcher only ever uses sample counts that are multiples of
// 16 (B = 16384, chunk &= ~15, so every cn is a multiple of 16). Hence all
// GEMM row counts (cn*81, cn*49, cn*25, cn) are multiples of 16 and every
// M-tile is full: no per-element row guards are needed, only the wave-uniform
// m0 >= M early-out for grid overshoot.
//
// Fragment layouts (ISA 7.12.2):
//   A (16x4, MxK): lane L: m = L&15; lanes<16 hold K0(v.x),K1(v.y);
//                  lanes>=16 hold K2(v.x),K3(v.y)
//   B (4x16, KxN): lane L: n = L&15; lanes<16 hold rows K0(v.x),K1(v.y);
//                  lanes>=16 hold rows K2(v.x),K3(v.y)
//   C/D (16x16):   vgpr i: lanes<16 -> (m0+i, n=lane); lanes>=16 -> (m0+8+i, lane-16)
// ---------------------------------------------------------------------------

typedef float v2f __attribute__((ext_vector_type(2)));
typedef float v8f __attribute__((ext_vector_type(8)));

#define WMMA4(a, b, c) \
  __builtin_amdgcn_wmma_f32_16x16x4_f32(false, (a), false, (b), (short)0, (c), false, false)

__device__ __forceinline__ float lrelu(float v) { return v >= 0.0f ? v : 0.01f * v; }

#define WAVE_SETUP(M)                                   \
  int lane = threadIdx.x & 31;                          \
  int wv   = threadIdx.x >> 5;                          \
  int m0   = (blockIdx.x * 8 + wv) * 16;                \
  if (m0 >= (M)) return;                                \
  int lm = lane & 15;                                   \
  int kb = (lane >> 4) * 2;                             \
  int mh = m0 + 8 * (lane >> 4);

// ---------------------------------------------------------------------------
// Weight prep: transpose to [K][N] (zero-padded). Run once per launch.
// ---------------------------------------------------------------------------
// Linear weight w[N][K] -> o[K][N]
__global__ void k_tr_lin(const float* __restrict__ w, float* __restrict__ o,
                         int N, int K) {
  int t = blockIdx.x * blockDim.x + threadIdx.x;
  if (t >= N * K) return;
  int k = t / N, n = t % N;
  o[t] = w[n * K + k];
}

// Conv weight w[N][Cin][3][3] -> o[Kpad][Npad], k = (ky*3+kx)*Cin + ci,
// zero outside (n < N, kk < 9).
__global__ void k_tr_conv(const float* __restrict__ w, float* __restrict__ o,
                          int N, int Npad, int Cin, int Kpad) {
  int t = blockIdx.x * blockDim.x + threadIdx.x;
  if (t >= Kpad * Npad) return;
  int k = t / Npad, n = t % Npad;
  int kk = k / Cin, ci = k % Cin;
  o[t] = (n < N && kk < 9) ? w[(n * Cin + ci) * 9 + kk] : 0.0f;
}

// ---------------------------------------------------------------------------
// K1: conv1  x[B,1,11,11] -> h[rows=cn*81, 64]   (im2col GEMM, K padded to 12)
// w1t: [12][64], rows 9..11 zero -> A side only needs address clamping.
// ---------------------------------------------------------------------------
__global__ __launch_bounds__(256) void k_conv1(
    const float* __restrict__ x, const float* __restrict__ w1t,
    const float* __restrict__ b1, float* __restrict__ h, int s0, int M) {
  WAVE_SETUP(M)
  int r = m0 + lm;
  int b = r / 81, pos = r % 81;
  int oy = pos / 9, ox = pos % 9;
  const float* xb = x + (size_t)(s0 + b) * 121 + oy * 11 + ox;

  v8f acc[4] = {};
  #pragma unroll
  for (int k0 = 0; k0 < 12; k0 += 4) {
    int k  = k0 + kb;
    int ka = (k     < 8) ? k     : 8;              // clamp: padded-K weight rows are 0
    int kc = (k + 1 < 8) ? k + 1 : 8;
    v2f a;
    a.x = xb[(ka / 3) * 11 + (ka % 3)];
    a.y = xb[(kc / 3) * 11 + (kc % 3)];
    #pragma unroll
    for (int nt = 0; nt < 4; ++nt) {
      int n = nt * 16 + lm;
      v2f bf;
      bf.x = w1t[(k    ) * 64 + n];
      bf.y = w1t[(k + 1) * 64 + n];
      acc[nt] = WMMA4(a, bf, acc[nt]);
    }
  }
  float bb[4];
  #pragma unroll
  for (int nt = 0; nt < 4; ++nt) bb[nt] = b1[nt * 16 + lm];
  #pragma unroll
  for (int i = 0; i < 8; ++i) {
    size_t row = (size_t)(mh + i) * 64 + lm;
    #pragma unroll
    for (int nt = 0; nt < 4; ++nt)
      h[row + nt * 16] = acc[nt][i] + bb[nt];       // conv1: no activation
  }
}

// ---------------------------------------------------------------------------
// K2: NLB theta/phi/g — three fused GEMMs (K=64, N=32); stores s=theta*phi, g
// wtt/wpt/wgt: [64][32]
// ---------------------------------------------------------------------------
__global__ __launch_bounds__(256) void k_nlb_tpg(
    const float* __restrict__ h,
    const float* __restrict__ wtt, const float* __restrict__ bt,
    const float* __restrict__ wpt, const float* __restrict__ bp,
    const float* __restrict__ wgt, const float* __restrict__ bg,
    float* __restrict__ s_out, float* __restrict__ g_out, int M) {
  WAVE_SETUP(M)
  const float* arow = h + (size_t)(m0 + lm) * 64;

  v8f at[2] = {}, ap[2] = {}, agc[2] = {};
  #pragma unroll 4
  for (int k0 = 0; k0 < 64; k0 += 4) {
    v2f a; a.x = arow[k0 + kb]; a.y = arow[k0 + kb + 1];
    #pragma unroll
    for (int nt = 0; nt < 2; ++nt) {
      int n = nt * 16 + lm;
      v2f bta, bpa, bga;
      bta.x = wtt[(k0 + kb) * 32 + n]; bta.y = wtt[(k0 + kb + 1) * 32 + n];
      bpa.x = wpt[(k0 + kb) * 32 + n]; bpa.y = wpt[(k0 + kb + 1) * 32 + n];
      bga.x = wgt[(k0 + kb) * 32 + n]; bga.y = wgt[(k0 + kb + 1) * 32 + n];
      at[nt]  = WMMA4(a, bta, at[nt]);
      ap[nt]  = WMMA4(a, bpa, ap[nt]);
      agc[nt] = WMMA4(a, bga, agc[nt]);
    }
  }
  float bbt[2], bbp[2], bbg[2];
  #pragma unroll
  for (int nt = 0; nt < 2; ++nt) {
    bbt[nt] = bt[nt * 16 + lm];
    bbp[nt] = bp[nt * 16 + lm];
    bbg[nt] = bg[nt * 16 + lm];
  }
  #pragma unroll
  for (int i = 0; i < 8; ++i) {
    size_t row = (size_t)(mh + i) * 32 + lm;
    #pragma unroll
    for (int nt = 0; nt < 2; ++nt) {
      float th = at[nt][i] + bbt[nt];
      float ph = ap[nt][i] + bbp[nt];
      s_out[row + nt * 16] = th * ph;
      g_out[row + nt * 16] = agc[nt][i] + bbg[nt];
    }
  }
}

// ---------------------------------------------------------------------------
// K3: softmax over W (9 values, stride 32) then multiply by g; in-place into s
// ---------------------------------------------------------------------------
__global__ void k_softmax_g(float* __restrict__ s, const float* __restrict__ g, int cn) {
  int t = blockIdx.x * blockDim.x + threadIdx.x;
  int total = cn * 9 * 32;
  if (t >= total) return;
  int ch = t & 31;
  int brow = t >> 5;                                 // b*9 + oy
  size_t base = (size_t)brow * 9 * 32 + ch;          // ((b*81 + oy*9 + ox)*32 + ch)
  float v[9], mx = -3.0e38f;
  #pragma unroll
  for (int ox = 0; ox < 9; ++ox) { v[ox] = s[base + (size_t)ox * 32]; mx = fmaxf(mx, v[ox]); }
  float sum = 0.0f;
  #pragma unroll
  for (int ox = 0; ox < 9; ++ox) { v[ox] = __expf(v[ox] - mx); sum += v[ox]; }
  float inv = 1.0f / sum;
  #pragma unroll
  for (int ox = 0; ox < 9; ++ox)
    s[base + (size_t)ox * 32] = v[ox] * inv * g[base + (size_t)ox * 32];
}

// ---------------------------------------------------------------------------
// K4: o = (attn*g) @ wo^T + bo + h, LeakyReLU, in-place into h  (K=32, N=64)
// wot: [32][64]
// ---------------------------------------------------------------------------
__global__ __launch_bounds__(256) void k_nlb_out(
    const float* __restrict__ ag, const float* __restrict__ wot,
    const float* __restrict__ bo, float* __restrict__ h, int M) {
  WAVE_SETUP(M)
  const float* arow = ag + (size_t)(m0 + lm) * 32;

  v8f acc[4] = {};
  #pragma unroll
  for (int k0 = 0; k0 < 32; k0 += 4) {
    v2f a; a.x = arow[k0 + kb]; a.y = arow[k0 + kb + 1];
    #pragma unroll
    for (int nt = 0; nt < 4; ++nt) {
      int n = nt * 16 + lm;
      v2f bf;
      bf.x = wot[(k0 + kb) * 64 + n];
      bf.y = wot[(k0 + kb + 1) * 64 + n];
      acc[nt] = WMMA4(a, bf, acc[nt]);
    }
  }
  float bb[4];
  #pragma unroll
  for (int nt = 0; nt < 4; ++nt) bb[nt] = bo[nt * 16 + lm];
  #pragma unroll
  for (int i = 0; i < 8; ++i) {
    size_t row = (size_t)(mh + i) * 64 + lm;
    #pragma unroll
    for (int nt = 0; nt < 4; ++nt) {
      float v = acc[nt][i] + bb[nt] + h[row + nt * 16];
      h[row + nt * 16] = lrelu(v);
    }
  }
}

// ---------------------------------------------------------------------------
// K5: conv2  h[cn*81,64] -> out[cn*49,32]  (gathered im2col GEMM, K=576)
// w2t: [576][32], k = (ky*3+kx)*64 + ci
// ---------------------------------------------------------------------------
__global__ __launch_bounds__(256) void k_conv2(
    const float* __restrict__ h, const float* __restrict__ w2t,
    const float* __restrict__ b2, float* __restrict__ out, int M) {
  WAVE_SETUP(M)
  int r = m0 + lm;
  int b = r / 49, p = r % 49, oy = p / 7, ox = p % 7;
  const float* abase = h + ((size_t)b * 81 + oy * 9 + ox) * 64;

  v8f acc[2] = {};
  for (int kk = 0; kk < 9; ++kk) {                   // (ky,kx)
    const float* arow = abase + ((kk / 3) * 9 + (kk % 3)) * 64;
    const float* wrow = w2t + kk * 64 * 32;
    #pragma unroll 4
    for (int c0 = 0; c0 < 64; c0 += 4) {
      v2f a; a.x = arow[c0 + kb]; a.y = arow[c0 + kb + 1];
      #pragma unroll
      for (int nt = 0; nt < 2; ++nt) {
        int n = nt * 16 + lm;
        v2f bf;
        bf.x = wrow[(c0 + kb) * 32 + n];
        bf.y = wrow[(c0 + kb + 1) * 32 + n];
        acc[nt] = WMMA4(a, bf, acc[nt]);
      }
    }
  }
  float bb[2];
  #pragma unroll
  for (int nt = 0; nt < 2; ++nt) bb[nt] = b2[nt * 16 + lm];
  #pragma unroll
  for (int i = 0; i < 8; ++i) {
    size_t row = (size_t)(mh + i) * 32 + lm;
    #pragma unroll
    for (int nt = 0; nt < 2; ++nt)
      out[row + nt * 16] = lrelu(acc[nt][i] + bb[nt]);
  }
}

// ---------------------------------------------------------------------------
// K6: conv3  in[cn*49,32] -> flat[cn,200] in NCHW-flatten order (c*25 + y*5 + x)
// w3t: [288][16], cols 8..15 zero -> unconditional B loads; store masks lm<8.
// ---------------------------------------------------------------------------
__global__ __launch_bounds__(256) void k_conv3(
    const float* __restrict__ in, const float* __restrict__ w3t,
    const float* __restrict__ b3, float* __restrict__ flat, int M) {
  WAVE_SETUP(M)
  int r = m0 + lm;
  int b = r / 25, p = r % 25, oy = p / 5, ox = p % 5;
  const float* abase = in + ((size_t)b * 49 + oy * 7 + ox) * 32;

  v8f acc = {};
  for (int kk = 0; kk < 9; ++kk) {
    const float* arow = abase + ((kk / 3) * 7 + (kk % 3)) * 32;
    const float* wrow = w3t + kk * 32 * 16;
    #pragma unroll 4
    for (int c0 = 0; c0 < 32; c0 += 4) {
      v2f a; a.x = arow[c0 + kb]; a.y = arow[c0 + kb + 1];
      v2f bf;
      bf.x = wrow[(c0 + kb) * 16 + lm];
      bf.y = wrow[(c0 + kb + 1) * 16 + lm];
      acc = WMMA4(a, bf, acc);
    }
  }
  float bias = b3[lm & 7];                           // unconditional; lanes >=8 unused
  bool nvalid = (lm < 8);
  #pragma unroll
  for (int i = 0; i < 8; ++i) {
    int m = mh + i;
    if (nvalid) {
      int bb = m / 25, pp = m % 25;
      flat[(size_t)bb * 200 + lm * 25 + pp] = lrelu(acc[i] + bias);
    }
  }
}

// ---------------------------------------------------------------------------
// K7: dense GEMM template: out[M,NT*16] = lrelu(A[M,K] @ Wt + bias), Wt [K][NT*16]
// ---------------------------------------------------------------------------
template <int K, int NT>
__global__ __launch_bounds__(256) void k_dense(
    const float* __restrict__ A, const float* __restrict__ Wt,
    const float* __restrict__ bias, float* __restrict__ out, int M) {
  WAVE_SETUP(M)
  const float* arow = A + (size_t)(m0 + lm) * K;

  v8f acc[NT] = {};
  #pragma unroll 2
  for (int k0 = 0; k0 < K; k0 += 4) {
    v2f a; a.x = arow[k0 + kb]; a.y = arow[k0 + kb + 1];
    #pragma unroll
    for (int nt = 0; nt < NT; ++nt) {
      int n = nt * 16 + lm;
      v2f bf;
      bf.x = Wt[(k0 + kb) * (NT * 16) + n];
      bf.y = Wt[(k0 + kb + 1) * (NT * 16) + n];
      acc[nt] = WMMA4(a, bf, acc[nt]);
    }
  }
  float bb[NT];
  #pragma unroll
  for (int nt = 0; nt < NT; ++nt) bb[nt] = bias[nt * 16 + lm];
  #pragma unroll
  for (int i = 0; i < 8; ++i) {
    size_t row = (size_t)(mh + i) * (NT * 16) + lm;
    #pragma unroll
    for (int nt = 0; nt < NT; ++nt)
      out[row + nt * 16] = lrelu(acc[nt][i] + bb[nt]);
  }
}

// ---------------------------------------------------------------------------
// K8: dense3 -> dense4 -> dense5 fused (tiny; VALU). Writes final output.
// ---------------------------------------------------------------------------
__global__ void k_dense_tail(
    const float* __restrict__ in,                        // [cn,32]
    const float* __restrict__ dw3, const float* __restrict__ db3,
    const float* __restrict__ dw4, const float* __restrict__ db4,
    const float* __restrict__ dw5, const float* __restrict__ db5,
    float* __restrict__ out, int s0, int cn) {
  int b = blockIdx.x * blockDim.x + threadIdx.x;
  if (b >= cn) return;
  float x2[32];
  #pragma unroll
  for (int i = 0; i < 32; ++i) x2[i] = in[(size_t)b * 32 + i];
  float x3[16];
  #pragma unroll
  for (int o = 0; o < 16; ++o) {
    float s = db3[o];
    #pragma unroll
    for (int i = 0; i < 32; ++i) s = fmaf(dw3[o * 32 + i], x2[i], s);
    x3[o] = lrelu(s);
  }
  float x4[8];
  #pragma unroll
  for (int o = 0; o < 8; ++o) {
    float s = db4[o];
    #pragma unroll
    for (int i = 0; i < 16; ++i) s = fmaf(dw4[o * 16 + i], x3[i], s);
    x4[o] = lrelu(s);
  }
  #pragma unroll
  for (int o = 0; o < 2; ++o) {
    float s = db5[o];
    #pragma unroll
    for (int i = 0; i < 8; ++i) s = fmaf(dw5[o * 8 + i], x4[i], s);
    out[(size_t)(s0 + b) * 2 + o] = s;
  }
}

// ---------------------------------------------------------------------------
extern "C" void kernel_launch(void* const* d_in, const int* in_sizes, int n_in,
                              void* d_out, int out_size, void* d_ws, size_t ws_size,
                              hipStream_t stream) {
  (void)in_sizes; (void)n_in; (void)out_size;
  const float* x   = (const float*)d_in[0];
  const float* w1  = (const float*)d_in[1];  const float* b1  = (const float*)d_in[2];
  const float* wt  = (const float*)d_in[3];  const float* bt  = (const float*)d_in[4];
  const float* wp  = (const float*)d_in[5];  const float* bp  = (const float*)d_in[6];
  const float* wg  = (const float*)d_in[7];  const float* bg  = (const float*)d_in[8];
  const float* wo  = (const float*)d_in[9];  const float* bo  = (const float*)d_in[10];
  const float* w2  = (const float*)d_in[11]; const float* b2  = (const float*)d_in[12];
  const float* w3  = (const float*)d_in[13]; const float* b3  = (const float*)d_in[14];
  const float* dw1 = (const float*)d_in[15]; const float* db1 = (const float*)d_in[16];
  const float* dw2 = (const float*)d_in[17]; const float* db2 = (const float*)d_in[18];
  const float* dw3 = (const float*)d_in[19]; const float* db3 = (const float*)d_in[20];
  const float* dw4 = (const float*)d_in[21]; const float* db4 = (const float*)d_in[22];
  const float* dw5 = (const float*)d_in[23]; const float* db5 = (const float*)d_in[24];
  float* out = (float*)d_out;

  const int B = 16384;

  // --- transposed-weight prefix in workspace ---
  float* ws   = (float*)d_ws;
  float* w1t  = ws;                 // [12][64]   = 768
  float* wtt  = w1t + 768;          // [64][32]   = 2048
  float* wpt  = wtt + 2048;         // [64][32]
  float* wgt  = wpt + 2048;         // [64][32]
  float* wot  = wgt + 2048;         // [32][64]   = 2048
  float* w2t  = wot + 2048;         // [576][32]  = 18432
  float* w3t  = w2t + 18432;        // [288][16]  = 4608
  float* dw1t = w3t + 4608;         // [200][64]  = 12800
  float* dw2t = dw1t + 12800;       // [64][32]   = 2048
  const size_t WPREFIX = 768 + 3 * 2048 + 2048 + 18432 + 4608 + 12800 + 2048; // 46848
  float* data = ws + WPREFIX;

  k_tr_conv<<<(12 * 64 + 255) / 256, 256, 0, stream>>>(w1, w1t, 64, 64, 1, 12);
  k_tr_lin<<<(64 * 32 + 255) / 256, 256, 0, stream>>>(wt, wtt, 32, 64);
  k_tr_lin<<<(64 * 32 + 255) / 256, 256, 0, stream>>>(wp, wpt, 32, 64);
  k_tr_lin<<<(64 * 32 + 255) / 256, 256, 0, stream>>>(wg, wgt, 32, 64);
  k_tr_lin<<<(32 * 64 + 255) / 256, 256, 0, stream>>>(wo, wot, 64, 32);
  k_tr_conv<<<(576 * 32 + 255) / 256, 256, 0, stream>>>(w2, w2t, 32, 32, 64, 576);
  k_tr_conv<<<(288 * 16 + 255) / 256, 256, 0, stream>>>(w3, w3t, 8, 16, 32, 288);
  k_tr_lin<<<(200 * 64 + 255) / 256, 256, 0, stream>>>(dw1, dw1t, 64, 200);
  k_tr_lin<<<(64 * 32 + 255) / 256, 256, 0, stream>>>(dw2, dw2t, 32, 64);

  // --- chunk sizing from remaining workspace ---
  // Per-sample scratch: h[81*64] + s[81*32] + g[81*32] floats = 41472 bytes.
  const size_t per_sample = (size_t)(81 * 64 + 81 * 32 + 81 * 32) * sizeof(float);
  size_t avail = ws_size > WPREFIX * sizeof(float) ? ws_size - WPREFIX * sizeof(float) : 0;
  int chunk = (int)(avail / per_sample);
  chunk &= ~15;                       // keep cn a multiple of 16 (full tiles)
  if (chunk < 16) chunk = 16;
  if (chunk > B) chunk = B;

  float* buf_h = data;                                 // [chunk*81, 64]
  float* buf_s = buf_h + (size_t)chunk * 81 * 64;      // [chunk*81, 32]
  float* buf_g = buf_s + (size_t)chunk * 81 * 32;      // [chunk*81, 32] / flat200 [chunk,200]

  for (int s0 = 0; s0 < B; s0 += chunk) {
    int cn = (B - s0 < chunk) ? (B - s0) : chunk;      // always a multiple of 16

    int M  = cn * 81;
    int bl = ((M + 15) / 16 + 7) / 8;
    k_conv1<<<bl, 256, 0, stream>>>(x, w1t, b1, buf_h, s0, M);
    k_nlb_tpg<<<bl, 256, 0, stream>>>(buf_h, wtt, bt, wpt, bp, wgt, bg, buf_s, buf_g, M);

    int t3 = cn * 9 * 32;
    k_softmax_g<<<(t3 + 255) / 256, 256, 0, stream>>>(buf_s, buf_g, cn);

    k_nlb_out<<<bl, 256, 0, stream>>>(buf_s, wot, bo, buf_h, M);

    int M2  = cn * 49;
    int bl2 = ((M2 + 15) / 16 + 7) / 8;
    k_conv2<<<bl2, 256, 0, stream>>>(buf_h, w2t, b2, buf_s, M2);   // buf_s: ag dead

    int M3  = cn * 25;
    int bl3 = ((M3 + 15) / 16 + 7) / 8;
    k_conv3<<<bl3, 256, 0, stream>>>(buf_s, w3t, b3, buf_g, M3);   // buf_g: g dead

    // dense head — buf_h is free after conv2 consumed it
    float* d1o = buf_h;                         // [cn,64]
    float* d2o = buf_h + (size_t)chunk * 64;    // [cn,32]
    int blm = ((cn + 15) / 16 + 7) / 8;
    k_dense<200, 4><<<blm, 256, 0, stream>>>(buf_g, dw1t, db1, d1o, cn);
    k_dense<64, 2><<<blm, 256, 0, stream>>>(d1o, dw2t, db2, d2o, cn);
    k_dense_tail<<<(cn + 255) / 256, 256, 0, stream>>>(
        d2o, dw3, db3, dw4, db4, dw5, db5, out, s0, cn);
  }
}